// OrdinalLoss_36386962932468
// MI455X (gfx1250) — compile-verified
//
#include <hip/hip_runtime.h>
#include <math.h>

#define K       101
#define BATCH   262144
#define TPB     256
#define WPB     8            // waves per block (wave32)
#define NBLOCKS 4096
#define ROWS_PER_WAVE (BATCH / (NBLOCKS * WPB))   // = 8

__device__ __forceinline__ float wave_sum(float v) {
#pragma unroll
  for (int d = 16; d >= 1; d >>= 1) v += __shfl_xor(v, d, 32);
  return v;
}
__device__ __forceinline__ float wave_max(float v) {
#pragma unroll
  for (int d = 16; d >= 1; d >>= 1) v = fmaxf(v, __shfl_xor(v, d, 32));
  return v;
}

// Issue one row (101 dwords) of async global->LDS copies (4 masked B32 issues).
// ldsoff is the byte offset of the destination buffer within the wave's LDS.
__device__ __forceinline__ void issue_row_async(const float* gptr, unsigned ldsoff, int lane) {
#pragma unroll
  for (int k = 0; k < 4; ++k) {
    int j = lane + 32 * k;
    if (j < K) {
      unsigned long long ga = (unsigned long long)(const void*)(gptr + j);
      unsigned la = ldsoff + (unsigned)(j * 4);
      asm volatile("global_load_async_to_lds_b32 %0, %1, off"
                   :: "v"(la), "v"(ga) : "memory");
    }
  }
}

__global__ void __launch_bounds__(TPB)
ordinal_loss_main(const float* __restrict__ logit,
                  const int*   __restrict__ labels,
                  const float* __restrict__ cw,
                  float* __restrict__ ws,
                  float* __restrict__ out,
                  int use_ws) {
  __shared__ __align__(16) float smem[WPB][2][128];  // double-buffered row staging per wave
  __shared__ float bsum[WPB];

  const int lane = threadIdx.x & 31;
  const int wav  = threadIdx.x >> 5;
  const int waveId = blockIdx.x * WPB + wav;
  const int r0 = waveId * ROWS_PER_WAVE;

  // generic LDS pointer: low 32 bits == LDS byte offset (flat aperture layout)
  const unsigned lds0 = (unsigned)(unsigned long long)(void*)&smem[wav][0][0];
  const unsigned lds1 = (unsigned)(unsigned long long)(void*)&smem[wav][1][0];

  issue_row_async(logit + (long long)r0 * K, lds0, lane);

  const float inv_logK = 1.0f / logf((float)K);
  const float inv_Km1  = 1.0f / (float)(K - 1);

  const int jb = lane * 4;          // this lane's first column (contiguous chunk)

  float accL = 0.0f;                // per-lane partial (reduced once at the end)
  float accU = 0.0f;                // wave-uniform partial (ce + local-peak terms)

#pragma unroll 1
  for (int i = 0; i < ROWS_PER_WAVE; ++i) {
    const int cur = i & 1;
    if (i + 1 < ROWS_PER_WAVE) {
      issue_row_async(logit + (long long)(r0 + i + 1) * K, cur ? lds0 : lds1, lane);
      asm volatile("s_wait_asynccnt 0x4" ::: "memory");  // current row's 4 loads done
    } else {
      asm volatile("s_wait_asynccnt 0x0" ::: "memory");
    }
    const float* buf = &smem[wav][cur][0];

    const int   y  = labels[r0 + i];
    const float wy = cw[y];

    // ---- one b128 LDS read: 4 contiguous columns per lane ----
    const float4 xq = *reinterpret_cast<const float4*>(buf + jb);
    float xv[4] = { xq.x, xq.y, xq.z, xq.w };

    // ---- softmax: max / exp / sum ----
    float mx = -3.0e38f;
#pragma unroll
    for (int e = 0; e < 4; ++e) {
      float v = (jb + e < K) ? xv[e] : -3.0e38f;
      xv[e] = v;
      mx = fmaxf(mx, v);
    }
    mx = wave_max(mx);

    float pv[4];
    float s = 0.0f;
#pragma unroll
    for (int e = 0; e < 4; ++e) {
      float ev = (jb + e < K) ? __expf(xv[e] - mx) : 0.0f;
      pv[e] = ev;
      s += ev;
    }
    s = wave_sum(s);
    const float invS = 1.0f / s;
#pragma unroll
    for (int e = 0; e < 4; ++e) pv[e] *= invS;
    const float logS = __logf(s);

    // ---- uniform-lane picks: x[y], p[y], p[y-1], p[y+1] (y is wave-uniform) ----
    auto sel4 = [](const float* a, int e) -> float {
      return (e == 0) ? a[0] : ((e == 1) ? a[1] : ((e == 2) ? a[2] : a[3]));
    };
    const float xy = __shfl(sel4(xv, y & 3), y >> 2, 32);
    const float py = __shfl(sel4(pv, y & 3), y >> 2, 32);
    const float pl = (y > 0)     ? __shfl(sel4(pv, (y - 1) & 3), (y - 1) >> 2, 32) : 0.0f;
    const float pr = (y < K - 1) ? __shfl(sel4(pv, (y + 1) & 3), (y + 1) >> 2, 32) : 0.0f;

    const float ce = -wy * (xy - mx - logS) * inv_logK;
    const float lp = fmaxf(fmaxf(pl, pr) - (py - 0.25f), 0.0f);
    accU += ce + 2.5f * lp;          // 0.5 * W_LPEAK = 2.5

    // ---- CDF: in-lane inclusive prefix + single wave scan of lane totals ----
    const float c0 = pv[0];
    const float c1 = c0 + pv[1];
    const float c2 = c1 + pv[2];
    const float c3 = c2 + pv[3];
    float inc = c3;
#pragma unroll
    for (int d = 1; d < 32; d <<= 1) {
      float u = __shfl_up(inc, d, 32);
      if (lane >= d) inc += u;
    }
    const float excl = inc - c3;
    const float cdf[4] = { excl + c0, excl + c1, excl + c2, excl + c3 };

    // ---- ESE / tail / EMD lane partials ----
    float ese = 0.0f, tail = 0.0f, emd = 0.0f, uni = 0.0f;
#pragma unroll
    for (int e = 0; e < 4; ++e) {
      int j = jb + e;
      if (j < K) {
        float ct = (j <= y) ? 1.0f : 0.0f;
        float df = cdf[e] - ct;
        emd += df * df;
        float dj = (float)(j - y);
        ese += pv[e] * dj * dj;
        if (fabsf(dj) > 1.5f) tail += pv[e];
      }
    }

    // ---- unimodal hinge: 3 in-lane pairs + 1 boundary shuffle ----
    const float nxt3 = __shfl(pv[0], (lane + 1) & 31, 32);
    const float nb[4] = { pv[1], pv[2], pv[3], nxt3 };
#pragma unroll
    for (int e = 0; e < 4; ++e) {
      int m = jb + e;
      if (m < K - 1) {
        float d = pv[e] - nb[e];
        uni += (m < y) ? fmaxf(d, 0.0f) : fmaxf(-d, 0.0f);
      }
    }

    // scale per-row, accumulate per-lane (lane reduction deferred to after the loop)
    // coefficients fold ALPHA=0.5: ese*1*wy/(K-1)^2, emd*1/K, tail*2.5, uni*0.75/(K-1)
    accL += wy * inv_Km1 * inv_Km1 * ese
          + (1.0f / (float)K) * emd
          + 2.5f * tail
          + 0.75f * inv_Km1 * uni;
  }

  // single cross-lane reduction for the whole wave's work
  const float waveTotal = wave_sum(accL) + accU;

  // ---- block reduction ----
  if (lane == 0) bsum[wav] = waveTotal;
  __syncthreads();
  if (threadIdx.x == 0) {
    float t = 0.0f;
#pragma unroll
    for (int w = 0; w < WPB; ++w) t += bsum[w];
    if (use_ws) ws[blockIdx.x] = t;
    else        atomicAdd(out, t * (1.0f / (float)BATCH));
  }
}

__global__ void __launch_bounds__(TPB) finalize_ws(const float* __restrict__ ws,
                                                   float* __restrict__ out) {
  __shared__ float red[TPB];
  float s = 0.0f;
  for (int i = threadIdx.x; i < NBLOCKS; i += TPB) s += ws[i];
  red[threadIdx.x] = s;
  __syncthreads();
  for (int st = TPB / 2; st > 0; st >>= 1) {
    if (threadIdx.x < st) red[threadIdx.x] += red[threadIdx.x + st];
    __syncthreads();
  }
  if (threadIdx.x == 0) out[0] = red[0] * (1.0f / (float)BATCH);
}

__global__ void zero_out(float* out) { out[0] = 0.0f; }

extern "C" void kernel_launch(void* const* d_in, const int* in_sizes, int n_in,
                              void* d_out, int out_size, void* d_ws, size_t ws_size,
                              hipStream_t stream) {
  const float* logit  = (const float*)d_in[0];
  const int*   labels = (const int*)d_in[1];
  const float* cw     = (const float*)d_in[2];
  float*       out    = (float*)d_out;

  const int use_ws = (ws_size >= (size_t)NBLOCKS * sizeof(float)) ? 1 : 0;
  if (!use_ws) zero_out<<<1, 1, 0, stream>>>(out);
  ordinal_loss_main<<<NBLOCKS, TPB, 0, stream>>>(logit, labels, cw,
                                                 (float*)d_ws, out, use_ws);
  if (use_ws) finalize_ws<<<1, TPB, 0, stream>>>((const float*)d_ws, out);
}